// GCRNN_59966333387150
// MI455X (gfx1250) — compile-verified
//
#include <hip/hip_runtime.h>
#include <hip/hip_bf16.h>

typedef float v2f __attribute__((ext_vector_type(2)));
typedef float v8f __attribute__((ext_vector_type(8)));

static __device__ __forceinline__ v8f wmma_f32_16x16x4(v2f a, v2f b, v8f c) {
  // D = A(16x4 f32) * B(4x16 f32) + C(16x16 f32)
  return __builtin_amdgcn_wmma_f32_16x16x4_f32(false, a, false, b, (short)0, c,
                                               false, false);
}

// ---------------- init: deg=1 (self loop), zero accumulators ----------------
__global__ __launch_bounds__(256) void k_init(float* deg, float* g1, float* out,
                                              int n) {
  int i = blockIdx.x * blockDim.x + threadIdx.x;
  if (i < n) deg[i] = 1.0f;
  if (i < n * 2) out[i] = 0.0f;
  if (i < n * 12) g1[i] = 0.0f;
}

__global__ __launch_bounds__(256) void k_deg(const int* __restrict__ dst,
                                             float* deg, int e) {
  int i = blockIdx.x * blockDim.x + threadIdx.x;
  if (i < e) atomicAdd(&deg[dst[i]], 1.0f);
}

__global__ __launch_bounds__(256) void k_rsqrt(float* deg, int n) {
  int i = blockIdx.x * blockDim.x + threadIdx.x;
  if (i < n) deg[i] = rsqrtf(deg[i]);
}

// ------- h1[N,12] = x[N,8] @ W1[8,12]  (WMMA f32, 8 waves/block) ------------
__global__ __launch_bounds__(256) void k_gemm1(const float* __restrict__ X,
                                               const float* __restrict__ W1,
                                               float* __restrict__ H1,
                                               int ntiles) {
  const int tile = blockIdx.x * 8 + (threadIdx.x >> 5);  // uniform per wave
  if (tile >= ntiles) return;                            // whole-wave exit
  const int lane = threadIdx.x & 31;
  const int half = lane >> 4, l15 = lane & 15;
  const int row = tile * 16 + l15;  // A: M = lane&15
  const int kb = half * 2;          // A/B VGPR K base for this half
  const int nn = l15 < 12 ? l15 : 0;
  const float msk = l15 < 12 ? 1.0f : 0.0f;
  v8f c = {};
#pragma unroll
  for (int k0 = 0; k0 < 8; k0 += 4) {
    v2f a, b;
    a.x = X[row * 8 + k0 + kb + 0];
    a.y = X[row * 8 + k0 + kb + 1];
    b.x = W1[(k0 + kb + 0) * 12 + nn] * msk;
    b.y = W1[(k0 + kb + 1) * 12 + nn] * msk;
    c = wmma_f32_16x16x4(a, b, c);
  }
  if (l15 < 12) {
#pragma unroll
    for (int r = 0; r < 8; ++r)  // D: M = r + 8*half, N = l15
      H1[(size_t)(tile * 16 + r + 8 * half) * 12 + l15] = c[r];
  }
}

// ---------------- conv1 edge scatter: g1[d] += h1[s] * dinv[s]*dinv[d] ------
__global__ __launch_bounds__(256) void k_edge12(
    const int* __restrict__ src, const int* __restrict__ dst,
    const float* __restrict__ dinv, const float* __restrict__ h,
    float* __restrict__ out, int e) {
  int i = blockIdx.x * blockDim.x + threadIdx.x;
  if (i >= e) return;
  int s = src[i], d = dst[i];
  float w = dinv[s] * dinv[d];
  const float4* hp = (const float4*)(h + (size_t)s * 12);  // 16B aligned
  float4 v0 = hp[0], v1 = hp[1], v2 = hp[2];
  float* o = out + (size_t)d * 12;
  atomicAdd(o + 0, v0.x * w);  atomicAdd(o + 1, v0.y * w);
  atomicAdd(o + 2, v0.z * w);  atomicAdd(o + 3, v0.w * w);
  atomicAdd(o + 4, v1.x * w);  atomicAdd(o + 5, v1.y * w);
  atomicAdd(o + 6, v1.z * w);  atomicAdd(o + 7, v1.w * w);
  atomicAdd(o + 8, v2.x * w);  atomicAdd(o + 9, v2.y * w);
  atomicAdd(o + 10, v2.z * w); atomicAdd(o + 11, v2.w * w);
}

__global__ __launch_bounds__(256) void k_self12(const float* __restrict__ dinv,
                                                const float* __restrict__ h,
                                                const float* __restrict__ b,
                                                float* __restrict__ out, int n) {
  int i = blockIdx.x * blockDim.x + threadIdx.x;
  if (i >= n * 12) return;
  int node = i / 12, ch = i % 12;
  float di = dinv[node];
  out[i] += h[i] * di * di + b[ch];
}

// ------- xw[N,16] = g1[N,12] @ W_ih^T + (b_ih + b_hh)  (WMMA f32) -----------
__global__ __launch_bounds__(256) void k_gemm_xw(const float* __restrict__ G,
                                                 const float* __restrict__ Wih,
                                                 const float* __restrict__ bih,
                                                 const float* __restrict__ bhh,
                                                 float* __restrict__ XW,
                                                 int ntiles) {
  const int tile = blockIdx.x * 8 + (threadIdx.x >> 5);
  if (tile >= ntiles) return;
  const int lane = threadIdx.x & 31;
  const int half = lane >> 4, l15 = lane & 15;
  const int row = tile * 16 + l15;
  const int kb = half * 2;
  v8f c = {};
#pragma unroll
  for (int k0 = 0; k0 < 12; k0 += 4) {
    v2f a, b;
    a.x = G[row * 12 + k0 + kb + 0];
    a.y = G[row * 12 + k0 + kb + 1];
    b.x = Wih[l15 * 12 + k0 + kb + 0];  // B[k][n] = W_ih[n][k]
    b.y = Wih[l15 * 12 + k0 + kb + 1];
    c = wmma_f32_16x16x4(a, b, c);
  }
  float bias = bih[l15] + bhh[l15];
#pragma unroll
  for (int r = 0; r < 8; ++r)
    XW[(size_t)(tile * 16 + r + 8 * half) * 16 + l15] = c[r] + bias;
}

// ------- serial RNN: h = tanh(xw_t + W_hh h), 1 wave, split-K halves -------
__global__ __launch_bounds__(32) void k_rnn(const float* __restrict__ XW,
                                            const float* __restrict__ Whh,
                                            float* __restrict__ HS, int n) {
  const int lane = threadIdx.x;
  const int j = lane & 15;
  const int kh = (lane >> 4) * 8;  // lanes 0-15: k=0..7, lanes 16-31: k=8..15
  float w[8];
#pragma unroll
  for (int k = 0; k < 8; ++k) w[k] = Whh[j * 16 + kh + k];
  float h = 0.0f;  // lane j (both halves) holds h_j
  const int BT = 16;
  for (int t0 = 0; t0 < n; t0 += BT) {
    float xv[BT];
#pragma unroll
    for (int i = 0; i < BT; ++i) xv[i] = XW[(size_t)(t0 + i) * 16 + j];
    if (t0 + BT < n)  // speculative prefetch of next batch -> global_prefetch
      __builtin_prefetch(&XW[(size_t)(t0 + BT) * 16 + j], 0, 0);
#pragma unroll
    for (int i = 0; i < BT; ++i) {
      float a0 = 0.0f, a1 = 0.0f;
#pragma unroll
      for (int k = 0; k < 8; k += 2) {
        a0 = fmaf(w[k], __shfl(h, kh + k, 32), a0);
        a1 = fmaf(w[k + 1], __shfl(h, kh + k + 1, 32), a1);
      }
      float p = a0 + a1;
      p += __shfl_xor(p, 16, 32);  // combine the two K-halves
      h = tanhf(p + xv[i]);        // identical in lane j and j+16
      if (lane < 16) HS[(size_t)(t0 + i) * 16 + j] = h;
    }
  }
}

// ---------------- h2[N,2] = hs[N,16] @ W2[16,2]  (WMMA f32) -----------------
__global__ __launch_bounds__(256) void k_gemm2(const float* __restrict__ HS,
                                               const float* __restrict__ W2,
                                               float* __restrict__ H2,
                                               int ntiles) {
  const int tile = blockIdx.x * 8 + (threadIdx.x >> 5);
  if (tile >= ntiles) return;
  const int lane = threadIdx.x & 31;
  const int half = lane >> 4, l15 = lane & 15;
  const int row = tile * 16 + l15;
  const int kb = half * 2;
  const int nn = l15 < 2 ? l15 : 0;
  const float msk = l15 < 2 ? 1.0f : 0.0f;
  v8f c = {};
#pragma unroll
  for (int k0 = 0; k0 < 16; k0 += 4) {
    v2f a, b;
    a.x = HS[row * 16 + k0 + kb + 0];
    a.y = HS[row * 16 + k0 + kb + 1];
    b.x = W2[(k0 + kb + 0) * 2 + nn] * msk;
    b.y = W2[(k0 + kb + 1) * 2 + nn] * msk;
    c = wmma_f32_16x16x4(a, b, c);
  }
  if (l15 < 2) {
#pragma unroll
    for (int r = 0; r < 8; ++r)
      H2[(size_t)(tile * 16 + r + 8 * half) * 2 + l15] = c[r];
  }
}

__global__ __launch_bounds__(256) void k_edge2(
    const int* __restrict__ src, const int* __restrict__ dst,
    const float* __restrict__ dinv, const float* __restrict__ h,
    float* __restrict__ out, int e) {
  int i = blockIdx.x * blockDim.x + threadIdx.x;
  if (i >= e) return;
  int s = src[i], d = dst[i];
  float w = dinv[s] * dinv[d];
  const float2* hp = (const float2*)(h + (size_t)s * 2);
  float2 v = hp[0];
  float* o = out + (size_t)d * 2;
  atomicAdd(o + 0, v.x * w);
  atomicAdd(o + 1, v.y * w);
}

__global__ __launch_bounds__(256) void k_self2(const float* __restrict__ dinv,
                                               const float* __restrict__ h,
                                               const float* __restrict__ b,
                                               float* __restrict__ out, int n) {
  int i = blockIdx.x * blockDim.x + threadIdx.x;
  if (i >= n * 2) return;
  int node = i >> 1, ch = i & 1;
  float di = dinv[node];
  out[i] += h[i] * di * di + b[ch];
}

extern "C" void kernel_launch(void* const* d_in, const int* in_sizes, int n_in,
                              void* d_out, int out_size, void* d_ws,
                              size_t ws_size, hipStream_t stream) {
  (void)n_in; (void)out_size; (void)ws_size;
  const float* x = (const float*)d_in[0];
  const int* ei = (const int*)d_in[1];
  const float* W1 = (const float*)d_in[2];
  const float* b1 = (const float*)d_in[3];
  const float* Wih = (const float*)d_in[4];
  const float* Whh = (const float*)d_in[5];
  const float* bih = (const float*)d_in[6];
  const float* bhh = (const float*)d_in[7];
  const float* W2 = (const float*)d_in[8];
  const float* b2 = (const float*)d_in[9];
  float* out = (float*)d_out;

  const int n = in_sizes[0] / 8;  // 100000 (divisible by 16)
  const int e = in_sizes[1] / 2;  // 6400000
  const int ntiles = n / 16;      // 6250
  const int gemmBlocks = (ntiles + 7) / 8;
  const int* src = ei;
  const int* dst = ei + e;

  float* ws = (float*)d_ws;  // ~23.6 MB of f32 scratch
  float* deg = ws;                  // n       (becomes dinv in-place)
  float* h1 = deg + n;              // n*12
  float* g1 = h1 + (size_t)n * 12;  // n*12
  float* xw = g1 + (size_t)n * 12;  // n*16
  float* hs = xw + (size_t)n * 16;  // n*16
  float* h2 = hs + (size_t)n * 16;  // n*2

  k_init<<<(n * 12 + 255) / 256, 256, 0, stream>>>(deg, g1, out, n);
  k_deg<<<(e + 255) / 256, 256, 0, stream>>>(dst, deg, e);
  k_rsqrt<<<(n + 255) / 256, 256, 0, stream>>>(deg, n);
  k_gemm1<<<gemmBlocks, 256, 0, stream>>>(x, W1, h1, ntiles);
  k_edge12<<<(e + 255) / 256, 256, 0, stream>>>(src, dst, deg, h1, g1, e);
  k_self12<<<(n * 12 + 255) / 256, 256, 0, stream>>>(deg, h1, b1, g1, n);
  k_gemm_xw<<<gemmBlocks, 256, 0, stream>>>(g1, Wih, bih, bhh, xw, ntiles);
  k_rnn<<<1, 32, 0, stream>>>(xw, Whh, hs, n);
  k_gemm2<<<gemmBlocks, 256, 0, stream>>>(hs, W2, h2, ntiles);
  k_edge2<<<(e + 255) / 256, 256, 0, stream>>>(src, dst, deg, h2, out, e);
  k_self2<<<(n * 2 + 255) / 256, 256, 0, stream>>>(deg, h2, b2, out, n);
}